// DemandAwareRS_35708358099011
// MI455X (gfx1250) — compile-verified
//
#include <hip/hip_runtime.h>
#include <hip/hip_bf16.h>
#include <math.h>

// ---- problem dims ----
#define Bn   128
#define Ln   50
#define Sn   100
#define Vn   100000
#define Cn   1000
#define Dn   64
#define NDn  4

typedef __attribute__((ext_vector_type(2))) float v2f;
typedef __attribute__((ext_vector_type(8))) float v8f;
typedef __attribute__((ext_vector_type(4))) unsigned int u32x4;
typedef __attribute__((ext_vector_type(8))) unsigned int u32x8;

// =====================================================================
// Kernel A: session_emb gather, write gnn (4 broadcast copies),
//           masked-mean graph_rep, last-item gather.
// grid = B blocks, 256 threads.
// =====================================================================
__global__ void kA_gather(const int* __restrict__ nodes,
                          const float* __restrict__ emb_i,
                          const int* __restrict__ mask_node,
                          const int* __restrict__ last_idx,
                          float* __restrict__ gnn_out,
                          float* __restrict__ graph_rep,
                          float* __restrict__ last)
{
    int b = blockIdx.x;
    int t = threadIdx.x;           // 0..255
    int d = t & 63;
    int lrow = t >> 6;             // 0..3

    __shared__ float acc[4][64];
    __shared__ float ms[4];

    float a = 0.f;
    float msum = 0.f;
    for (int l = lrow; l < Ln; l += 4) {
        int node = nodes[b * Ln + l];
        float e = emb_i[(size_t)node * Dn + d];
        float m = (float)mask_node[b * Ln + l];
        a += m * e;
        if (d == 0) msum += m;
        size_t base = (size_t)b * (NDn * Ln * Dn) + (size_t)l * Dn + d;
        gnn_out[base]               = e;
        gnn_out[base + 1 * Ln * Dn] = e;
        gnn_out[base + 2 * Ln * Dn] = e;
        gnn_out[base + 3 * Ln * Dn] = e;
    }
    acc[lrow][d] = a;
    if (d == 0) ms[lrow] = msum;
    __syncthreads();

    if (t < 64) {
        float s = acc[0][t] + acc[1][t] + acc[2][t] + acc[3][t];
        float tm = ms[0] + ms[1] + ms[2] + ms[3];
        graph_rep[b * Dn + t] = s / tm;
        int li = last_idx[b];
        last[b * Dn + t] = emb_i[(size_t)nodes[b * Ln + li] * Dn + t];
    }
}

// =====================================================================
// Kernel D: candidate demand weight  wsum[v] = sum_n softmax_n(...)
//   t = tanh(emb_c[cc[v]] @ Wd1 + bd1)   (WMMA f32 16x16x4, K=64, N=64)
//   z = t @ Wd2 + bd2 ; wsum = sum(softmax(z))
// grid = ceil(V/128), 256 threads (8 waves, 16 candidates per wave).
// =====================================================================
__global__ void kD_wsum(const int* __restrict__ cand_cat,
                        const float* __restrict__ emb_c,
                        const float* __restrict__ Wd1,
                        const float* __restrict__ bd1,
                        const float* __restrict__ Wd2,
                        const float* __restrict__ bd2,
                        float* __restrict__ wsum)
{
    __shared__ float tile[8][16 * 64];   // per-wave tanh output tile

    int wave = threadIdx.x >> 5;
    int lane = threadIdx.x & 31;
    int hi   = lane >> 4;
    int lm   = lane & 15;

    int v0 = blockIdx.x * 128 + wave * 16;

    // A row = emb_c[cand_cat[v0+lm]]
    int vl = v0 + lm; if (vl >= Vn) vl = Vn - 1;
    const float* arow = emb_c + (size_t)cand_cat[vl] * Dn;

    v8f c0 = {}, c1 = {}, c2 = {}, c3 = {};
#pragma unroll
    for (int k0 = 0; k0 < 64; k0 += 4) {
        int k = k0 + 2 * hi;
        v2f a; a.x = arow[k]; a.y = arow[k + 1];
        v2f b0, b1, b2, b3;
        b0.x = Wd1[k * 64 + ( 0 + lm)]; b0.y = Wd1[(k + 1) * 64 + ( 0 + lm)];
        b1.x = Wd1[k * 64 + (16 + lm)]; b1.y = Wd1[(k + 1) * 64 + (16 + lm)];
        b2.x = Wd1[k * 64 + (32 + lm)]; b2.y = Wd1[(k + 1) * 64 + (32 + lm)];
        b3.x = Wd1[k * 64 + (48 + lm)]; b3.y = Wd1[(k + 1) * 64 + (48 + lm)];
        c0 = __builtin_amdgcn_wmma_f32_16x16x4_f32(false, a, false, b0, (short)0, c0, false, false);
        c1 = __builtin_amdgcn_wmma_f32_16x16x4_f32(false, a, false, b1, (short)0, c1, false, false);
        c2 = __builtin_amdgcn_wmma_f32_16x16x4_f32(false, a, false, b2, (short)0, c2, false, false);
        c3 = __builtin_amdgcn_wmma_f32_16x16x4_f32(false, a, false, b3, (short)0, c3, false, false);
    }
#pragma unroll
    for (int r = 0; r < 8; r++) {
        int row = r + 8 * hi;
        tile[wave][row * 64 + ( 0 + lm)] = tanhf(c0[r] + bd1[ 0 + lm]);
        tile[wave][row * 64 + (16 + lm)] = tanhf(c1[r] + bd1[16 + lm]);
        tile[wave][row * 64 + (32 + lm)] = tanhf(c2[r] + bd1[32 + lm]);
        tile[wave][row * 64 + (48 + lm)] = tanhf(c3[r] + bd1[48 + lm]);
    }
    __syncthreads();

    if (hi == 0) {                       // lanes 0..15: one candidate each
        int v = v0 + lm;
        if (v < Vn) {
            float z[NDn];
#pragma unroll
            for (int j = 0; j < NDn; j++) z[j] = bd2[j];
            for (int n = 0; n < 64; n++) {
                float tv = tile[wave][lm * 64 + n];
#pragma unroll
                for (int j = 0; j < NDn; j++) z[j] += tv * Wd2[n * NDn + j];
            }
            float mx = z[0];
#pragma unroll
            for (int j = 1; j < NDn; j++) mx = fmaxf(mx, z[j]);
            float den = 0.f;
#pragma unroll
            for (int j = 0; j < NDn; j++) { z[j] = expf(z[j] - mx); den += z[j]; }
            float s = 0.f;
#pragma unroll
            for (int j = 0; j < NDn; j++) s += z[j] / den;
            wsum[v] = s;                 // == 1.0 up to rounding, computed faithfully
        }
    }
}

// =====================================================================
// Kernel B: three small GEMMs with WMMA f32 16x16x4:
//   w    = graph_rep        @ W_im^T          (M=128,N=64,K=64)
//   wneg = roll(graph_rep)  @ W_im^T
//   h    = tanh([graph_rep,last] @ W_pvsd + b) (M=128,N=64,K=128)
// grid = 8 (M-tiles), 384 threads (12 waves: 4 N-tiles x 3 jobs).
// =====================================================================
__global__ void kB_small(const float* __restrict__ graph_rep,
                         const float* __restrict__ last,
                         const float* __restrict__ W_im,
                         const float* __restrict__ W_pvsd,
                         const float* __restrict__ b_pvsd,
                         float* __restrict__ w,
                         float* __restrict__ wneg,
                         float* __restrict__ h)
{
    int mt    = blockIdx.x;            // 0..7
    int wave  = threadIdx.x >> 5;      // 0..11
    int lane  = threadIdx.x & 31;
    int which = wave >> 2;             // 0: w, 1: wneg, 2: h
    int nt    = wave & 3;
    int hi    = lane >> 4;
    int lm    = lane & 15;
    int n     = nt * 16 + lm;
    int m     = mt * 16 + lm;

    v8f c = {};
    if (which < 2) {
        int arow = (which == 1) ? ((m + Bn - 1) & (Bn - 1)) : m;  // roll by 1
#pragma unroll
        for (int k0 = 0; k0 < 64; k0 += 4) {
            int k = k0 + 2 * hi;
            v2f a, b;
            a.x = graph_rep[arow * Dn + k];
            a.y = graph_rep[arow * Dn + k + 1];
            b.x = W_im[n * Dn + k];        // B[k][n] = W_im[n][k]
            b.y = W_im[n * Dn + k + 1];
            c = __builtin_amdgcn_wmma_f32_16x16x4_f32(false, a, false, b, (short)0, c, false, false);
        }
        float* dst = (which == 1) ? wneg : w;
#pragma unroll
        for (int r = 0; r < 8; r++) {
            int row = mt * 16 + r + 8 * hi;
            dst[row * Dn + n] = c[r];
        }
    } else {
#pragma unroll
        for (int k0 = 0; k0 < 128; k0 += 4) {
            int k = k0 + 2 * hi;
            const float* Asrc = (k < 64) ? graph_rep : last;
            int kk = k & 63;
            v2f a, b;
            a.x = Asrc[m * Dn + kk];
            a.y = Asrc[m * Dn + kk + 1];
            b.x = W_pvsd[k * Dn + n];
            b.y = W_pvsd[(k + 1) * Dn + n];
            c = __builtin_amdgcn_wmma_f32_16x16x4_f32(false, a, false, b, (short)0, c, false, false);
        }
#pragma unroll
        for (int r = 0; r < 8; r++) {
            int row = mt * 16 + r + 8 * hi;
            h[row * Dn + n] = tanhf(c[r] + b_pvsd[n]);
        }
    }
}

// =====================================================================
// Kernel C: infomax loss reduction. Single block (deterministic).
// =====================================================================
__device__ __forceinline__ float logsig(float x) {
    return fminf(x, 0.f) - log1pf(expf(-fabsf(x)));
}

__global__ void kC_loss(const float* __restrict__ gnn,   // (B,ND,L,D), use n=0
                        const float* __restrict__ w,
                        const float* __restrict__ wneg,
                        float* __restrict__ loss_out)
{
    __shared__ float red[256];
    float s = 0.f;
    for (int idx = threadIdx.x; idx < Bn * Ln; idx += 256) {
        int b = idx / Ln, l = idx % Ln;
        const float* se = gnn + (size_t)b * (NDn * Ln * Dn) + (size_t)l * Dn;
        const float* wb = w + b * Dn;
        const float* wn = wneg + b * Dn;
        float pos = 0.f, neg = 0.f;
#pragma unroll 8
        for (int d = 0; d < Dn; d++) {
            float e = se[d];
            pos = fmaf(e, wb[d], pos);
            neg = fmaf(e, wn[d], neg);
        }
        s += logsig(pos) + logsig(-neg);
    }
    red[threadIdx.x] = s;
    __syncthreads();
    for (int off = 128; off > 0; off >>= 1) {
        if (threadIdx.x < off) red[threadIdx.x] += red[threadIdx.x + off];
        __syncthreads();
    }
    if (threadIdx.x == 0) loss_out[0] = -(red[0] / (float)(Bn * Ln));
}

// =====================================================================
// Kernel E (dominant, bandwidth-bound): P_v = (h @ emb_i^T) * wsum
//   M=128, N=100000, K=64. Per block: 64-candidate column. The 16 KB
//   emb_i tile is DMA'd into LDS by the Tensor Data Mover (TDM), then
//   8 waves each cover one M-tile x 4 N-subtiles with WMMA f32 16x16x4.
// grid = ceil(V/64) = 1563, 256 threads.
// =====================================================================
__global__ void kE_pv(const float* __restrict__ h,
                      const float* __restrict__ emb_i,
                      const float* __restrict__ wsum,
                      float* __restrict__ Pv)
{
    __shared__ float ldsB[64 * 64];     // 16 KB: B[k][n] = ldsB[n*64+k]

    int v0 = blockIdx.x * 64;
    int rows = Vn - v0; if (rows > 64) rows = 64;   // ragged tail: TDM zero-fills OOB

    if (threadIdx.x < 32) {             // wave 0 issues the TDM transfer
        unsigned long long gaddr =
            (unsigned long long)(uintptr_t)emb_i + (unsigned long long)v0 * (Dn * 4);
        unsigned int lds_off = (unsigned int)(uintptr_t)(void*)ldsB;  // LDS byte offset

        // D# group 0 (ISA 08 §8.3): count=1 | lds_addr | global_addr | type=2
        u32x4 g0;
        g0.x = 1u;
        g0.y = lds_off;
        g0.z = (unsigned int)gaddr;
        g0.w = (unsigned int)(gaddr >> 32) | (2u << 30);

        // D# group 1 (§8.4): data_size=4B, 2D tile 64x64 elems, stride 64
        u32x8 g1;
        g1[0] = 0x00020000u;                       // data_size=2 (4 bytes)
        g1[1] = 64u << 16;                         // tensor_dim0 = 64
        g1[2] = ((unsigned int)rows) << 16;        // tensor_dim1 = valid rows
        g1[3] = 64u << 16;                         // tile_dim0 = 64
        g1[4] = 64u;                               // tile_dim1 = 64, tile_dim2 = 0
        g1[5] = 64u;                               // tensor_dim0_stride = 64
        g1[6] = 0u;
        g1[7] = 0u;

        asm volatile("tensor_load_to_lds %0, %1" :: "s"(g0), "s"(g1) : "memory");
        __builtin_amdgcn_s_wait_tensorcnt(0);
    }
    __syncthreads();

    int wave = threadIdx.x >> 5;        // M-tile 0..7
    int lane = threadIdx.x & 31;
    int hi   = lane >> 4;
    int lm   = lane & 15;
    int mrow = wave * 16 + lm;

    __builtin_prefetch(h + mrow * Dn, 0, 0);       // global_prefetch_b8

    v8f c[4] = {{}, {}, {}, {}};
#pragma unroll
    for (int k0 = 0; k0 < 64; k0 += 4) {
        int k = k0 + 2 * hi;
        v2f a;                                     // A fragment reused across N
        a.x = h[mrow * Dn + k];
        a.y = h[mrow * Dn + k + 1];
#pragma unroll
        for (int nt = 0; nt < 4; nt++) {
            v2f b;
            b.x = ldsB[(nt * 16 + lm) * 64 + k];
            b.y = ldsB[(nt * 16 + lm) * 64 + k + 1];
            c[nt] = __builtin_amdgcn_wmma_f32_16x16x4_f32(false, a, false, b,
                                                          (short)0, c[nt], false, false);
        }
    }

#pragma unroll
    for (int nt = 0; nt < 4; nt++) {
        int v = v0 + nt * 16 + lm;
        if (v < Vn) {
            float wsc = wsum[v];
#pragma unroll
            for (int r = 0; r < 8; r++) {
                int row = wave * 16 + r + 8 * hi;  // batch index
                Pv[(size_t)row * Vn + v] = c[nt][r] * wsc;
            }
        }
    }
}

// =====================================================================
extern "C" void kernel_launch(void* const* d_in, const int* in_sizes, int n_in,
                              void* d_out, int out_size, void* d_ws, size_t ws_size,
                              hipStream_t stream) {
    const int*   nodes     = (const int*)d_in[0];
    // d_in[1] categories, d_in[2] adj, d_in[3] nodes_categories: unused in outputs
    const int*   last_idx  = (const int*)d_in[4];
    const int*   cand_cat  = (const int*)d_in[5];
    const int*   mask_node = (const int*)d_in[6];
    const float* emb_i     = (const float*)d_in[7];
    const float* emb_c     = (const float*)d_in[8];
    const float* Wd1       = (const float*)d_in[9];
    const float* bd1       = (const float*)d_in[10];
    const float* Wd2       = (const float*)d_in[11];
    const float* bd2       = (const float*)d_in[12];
    const float* W_pvsd    = (const float*)d_in[13];
    const float* b_pvsd    = (const float*)d_in[14];
    const float* W_im      = (const float*)d_in[15];

    float* Pv   = (float*)d_out;                       // (B, V)
    float* loss = Pv + (size_t)Bn * Vn;                // scalar
    float* gnn  = loss + 1;                            // (B, ND, L, D)

    float* ws        = (float*)d_ws;
    float* graph_rep = ws;                             // 128*64
    float* last      = ws + 8192;                      // 128*64
    float* wv        = ws + 16384;                     // 128*64
    float* wneg      = ws + 24576;                     // 128*64
    float* hb        = ws + 32768;                     // 128*64
    float* wsum      = ws + 40960;                     // 100000

    kA_gather<<<Bn, 256, 0, stream>>>(nodes, emb_i, mask_node, last_idx,
                                      gnn, graph_rep, last);
    kD_wsum<<<(Vn + 127) / 128, 256, 0, stream>>>(cand_cat, emb_c, Wd1, bd1,
                                                  Wd2, bd2, wsum);
    kB_small<<<8, 384, 0, stream>>>(graph_rep, last, W_im, W_pvsd, b_pvsd,
                                    wv, wneg, hb);
    kC_loss<<<1, 256, 0, stream>>>(gnn, wv, wneg, loss);
    kE_pv<<<(Vn + 63) / 64, 256, 0, stream>>>(hb, emb_i, wsum, Pv);
}